// dynGATtransformer_83141976916904
// MI455X (gfx1250) — compile-verified
//
#include <hip/hip_runtime.h>
#include <hip/hip_bf16.h>

// Sizes from the reference
#define NN   512
#define DD   256
#define HH   8
#define CCd  256
#define HCc  2048
#define EEe  7680
#define HD32 32
#define GGg  32
#define EPG  240

typedef float v2f __attribute__((ext_vector_type(2)));
typedef float v8f __attribute__((ext_vector_type(8)));

// ---------------- block reductions ----------------
__device__ __forceinline__ float block_sum(float v, float* sh) {
  int t = threadIdx.x;
  sh[t] = v; __syncthreads();
  for (int s = blockDim.x >> 1; s > 0; s >>= 1) {
    if (t < s) sh[t] += sh[t + s];
    __syncthreads();
  }
  float r = sh[0]; __syncthreads();
  return r;
}
__device__ __forceinline__ float block_max(float v, float* sh) {
  int t = threadIdx.x;
  sh[t] = v; __syncthreads();
  for (int s = blockDim.x >> 1; s > 0; s >>= 1) {
    if (t < s) sh[t] = fmaxf(sh[t], sh[t + s]);
    __syncthreads();
  }
  float r = sh[0]; __syncthreads();
  return r;
}

// ---------------- WMMA GEMMs (fp32, V_WMMA_F32_16X16X4_F32) ----------------
// One wave computes a 32x32 output tile (2x2 grid of 16x16 WMMA tiles):
// 4 fragment loads feed 4 WMMAs per k-step (2x the flops/byte of 16x16 tiling).
// All M / Nc in this net are multiples of 32, so no remainder paths.

// out[m][n] = alpha * sum_k A[m][k]*B[n][k] + bias[n]   (B is Nc x K row-major)
// grid = (Nc/32, M/32, batch)
__global__ __launch_bounds__(32)
void gemm_nt_wmma(const float* __restrict__ A, const float* __restrict__ B,
                  const float* __restrict__ bias, float* __restrict__ out,
                  int K, int lda, int ldb, int ldc, float alpha,
                  long sA, long sB, long sO)
{
  long bz = blockIdx.z;
  A += bz * sA; B += bz * sB; out += bz * sO;
  const int lane  = threadIdx.x;
  const int r     = lane & 15;
  const int kh    = (lane >> 4) << 1;          // 0 or 2 (K sub-offset per lane half)
  const int tileN = blockIdx.x << 5;
  const int tileM = blockIdx.y << 5;
  const float* Ap0 = A + (long)(tileM + r) * lda + kh;
  const float* Ap1 = Ap0 + 16L * lda;
  const float* Bp0 = B + (long)(tileN + r) * ldb + kh;
  const float* Bp1 = Bp0 + 16L * ldb;
  v8f acc00 = {0.f,0.f,0.f,0.f,0.f,0.f,0.f,0.f};
  v8f acc01 = acc00, acc10 = acc00, acc11 = acc00;
  #pragma unroll 2
  for (int k = 0; k < K; k += 4) {
    v2f a0, a1, b0, b1;
    a0.x = Ap0[k]; a0.y = Ap0[k + 1];
    a1.x = Ap1[k]; a1.y = Ap1[k + 1];
    b0.x = Bp0[k]; b0.y = Bp0[k + 1];
    b1.x = Bp1[k]; b1.y = Bp1[k + 1];
    acc00 = __builtin_amdgcn_wmma_f32_16x16x4_f32(false, a0, false, b0, (short)0, acc00, false, false);
    acc01 = __builtin_amdgcn_wmma_f32_16x16x4_f32(false, a0, false, b1, (short)0, acc01, false, false);
    acc10 = __builtin_amdgcn_wmma_f32_16x16x4_f32(false, a1, false, b0, (short)0, acc10, false, false);
    acc11 = __builtin_amdgcn_wmma_f32_16x16x4_f32(false, a1, false, b1, (short)0, acc11, false, false);
  }
  const int m0   = (lane >> 4) << 3;           // 0 or 8 (C/D row half)
  const int col0 = tileN + r, col1 = col0 + 16;
  const float bv0 = bias ? bias[col0] : 0.0f;
  const float bv1 = bias ? bias[col1] : 0.0f;
  #pragma unroll
  for (int v = 0; v < 8; ++v) {
    long row0 = tileM + m0 + v, row1 = row0 + 16;
    out[row0 * ldc + col0] = alpha * acc00[v] + bv0;
    out[row0 * ldc + col1] = alpha * acc01[v] + bv1;
    out[row1 * ldc + col0] = alpha * acc10[v] + bv0;
    out[row1 * ldc + col1] = alpha * acc11[v] + bv1;
  }
}

// out[m][n] = alpha * sum_k A[m][k]*B[k][n]   (B is K x Nc row-major)
// grid = (Nc/32, M/32, batch)
__global__ __launch_bounds__(32)
void gemm_nn_wmma(const float* __restrict__ A, const float* __restrict__ B,
                  const float* __restrict__ bias, float* __restrict__ out,
                  int K, int lda, int ldb, int ldc, float alpha,
                  long sA, long sB, long sO)
{
  long bz = blockIdx.z;
  A += bz * sA; B += bz * sB; out += bz * sO;
  const int lane  = threadIdx.x;
  const int r     = lane & 15;
  const int kh    = (lane >> 4) << 1;
  const int tileN = blockIdx.x << 5;
  const int tileM = blockIdx.y << 5;
  const float* Ap0 = A + (long)(tileM + r) * lda + kh;
  const float* Ap1 = Ap0 + 16L * lda;
  const float* Bp0 = B + (long)kh * ldb + (tileN + r);
  const float* Bp1 = Bp0 + 16;
  v8f acc00 = {0.f,0.f,0.f,0.f,0.f,0.f,0.f,0.f};
  v8f acc01 = acc00, acc10 = acc00, acc11 = acc00;
  #pragma unroll 2
  for (int k = 0; k < K; k += 4) {
    v2f a0, a1, b0, b1;
    long bo = (long)k * ldb;
    a0.x = Ap0[k];        a0.y = Ap0[k + 1];
    a1.x = Ap1[k];        a1.y = Ap1[k + 1];
    b0.x = Bp0[bo];       b0.y = Bp0[bo + ldb];
    b1.x = Bp1[bo];       b1.y = Bp1[bo + ldb];
    acc00 = __builtin_amdgcn_wmma_f32_16x16x4_f32(false, a0, false, b0, (short)0, acc00, false, false);
    acc01 = __builtin_amdgcn_wmma_f32_16x16x4_f32(false, a0, false, b1, (short)0, acc01, false, false);
    acc10 = __builtin_amdgcn_wmma_f32_16x16x4_f32(false, a1, false, b0, (short)0, acc10, false, false);
    acc11 = __builtin_amdgcn_wmma_f32_16x16x4_f32(false, a1, false, b1, (short)0, acc11, false, false);
  }
  const int m0   = (lane >> 4) << 3;
  const int col0 = tileN + r, col1 = col0 + 16;
  const float bv0 = bias ? bias[col0] : 0.0f;
  const float bv1 = bias ? bias[col1] : 0.0f;
  #pragma unroll
  for (int v = 0; v < 8; ++v) {
    long row0 = tileM + m0 + v, row1 = row0 + 16;
    out[row0 * ldc + col0] = alpha * acc00[v] + bv0;
    out[row0 * ldc + col1] = alpha * acc01[v] + bv1;
    out[row1 * ldc + col0] = alpha * acc10[v] + bv0;
    out[row1 * ldc + col1] = alpha * acc11[v] + bv1;
  }
}

// ---------------- embed + LN + pe ----------------
__global__ void embed_ln_pe(const float* __restrict__ x, const float* __restrict__ w,
                            const float* __restrict__ b, const float* __restrict__ g,
                            const float* __restrict__ bb, const float* __restrict__ pe,
                            float* __restrict__ h)
{
  int n = blockIdx.x, d = threadIdx.x;
  float acc = b[d];
  #pragma unroll
  for (int k = 0; k < 16; ++k) acc += x[n * 16 + k] * w[d * 16 + k];
  __shared__ float sh[256];
  float mean = block_sum(acc, sh) * (1.0f / DD);
  float dv = acc - mean;
  float var = block_sum(dv * dv, sh) * (1.0f / DD);
  h[n * DD + d] = dv * rsqrtf(var + 1e-5f) * g[d] + bb[d] + pe[n * DD + d];
}

// ---------------- GAT edge logits: one block per (edge, head) ----------------
__global__ void gat_edge_logits(const float* __restrict__ xl, const float* __restrict__ xr,
                                const float* __restrict__ ea, const float* __restrict__ ew,
                                const float* __restrict__ att, float* __restrict__ glogit)
{
  int e = blockIdx.x, hh = blockIdx.y, c = threadIdx.x;
  int g = e / EPG, rr = e % EPG;
  int s = rr / 15, bt = rr % 15;
  int t = bt + (bt >= s ? 1 : 0);
  int src = g * 16 + s, dst = g * 16 + t;
  int hc = hh * CCd + c;
  float te = 0.f;
  #pragma unroll
  for (int j = 0; j < 6; ++j) te += ea[e * 6 + j] * ew[hc * 6 + j];
  float v = xl[(long)src * HCc + hc] + xr[(long)dst * HCc + hc] + te;
  v = v > 0.f ? v : 0.2f * v;
  v *= att[hc];
  __shared__ float sh[256];
  float tot = block_sum(v, sh);
  if (c == 0) glogit[e * HH + hh] = tot;
}

// ---------------- GAT aggregate + LN: one block per dst node ----------------
__global__ void gat_aggregate(const float* __restrict__ xl, const float* __restrict__ glogit,
                              const float* __restrict__ gbias, const float* __restrict__ lng,
                              const float* __restrict__ lnb, float* __restrict__ xgat)
{
  int n = blockIdx.x, c = threadIdx.x;
  int g = n / 16, t = n % 16;
  __shared__ float w[HH][15];
  if (c < 120) {
    int si = c % 15, hh = c / 15;
    int s = si + (si >= t ? 1 : 0);
    int e = g * EPG + s * 15 + (s < t ? t - 1 : t);
    w[hh][si] = glogit[e * HH + hh];
  }
  __syncthreads();
  if (c < HH) {
    float m = -1e30f;
    for (int i = 0; i < 15; ++i) m = fmaxf(m, w[c][i]);
    float ssum = 0.f;
    for (int i = 0; i < 15; ++i) { float ev = __expf(w[c][i] - m); w[c][i] = ev; ssum += ev; }
    float inv = 1.0f / fmaxf(ssum, 1e-16f);
    for (int i = 0; i < 15; ++i) w[c][i] *= inv;
  }
  __syncthreads();
  float acc = 0.f;
  for (int si = 0; si < 15; ++si) {
    int s = si + (si >= t ? 1 : 0);
    long src = g * 16 + s;
    #pragma unroll
    for (int hh = 0; hh < HH; ++hh)
      acc += xl[src * HCc + hh * CCd + c] * w[hh][si];
  }
  acc = acc * 0.125f + gbias[c];
  __shared__ float sh[256];
  float mean = block_sum(acc, sh) * (1.0f / DD);
  float dv = acc - mean;
  float var = block_sum(dv * dv, sh) * (1.0f / DD);
  xgat[n * DD + c] = dv * rsqrtf(var + 1e-5f) * lng[c] + lnb[c];
}

// ---------------- transformer edge logits: one block per (node, head) ----------------
__global__ void trans_logits(const float* __restrict__ qt, const float* __restrict__ kt,
                             const float* __restrict__ ea, const float* __restrict__ tew,
                             float* __restrict__ l2)
{
  int n = blockIdx.x, hh = blockIdx.y, c = threadIdx.x;
  int g = n / 16, t = n % 16;
  int hc = hh * CCd + c;
  float q = qt[(long)n * HCc + hc];
  __shared__ float sh[256];
  for (int si = 0; si < 15; ++si) {
    int s = si + (si >= t ? 1 : 0);
    long src = g * 16 + s;
    int e = g * EPG + s * 15 + (s < t ? t - 1 : t);
    float te = 0.f;
    #pragma unroll
    for (int j = 0; j < 6; ++j) te += ea[e * 6 + j] * tew[hc * 6 + j];
    float kv = kt[src * HCc + hc] + te;
    float tot = block_sum(q * kv, sh);
    if (c == 0) l2[((long)n * HH + hh) * 16 + si] = tot * 0.0625f; // / sqrt(256)
  }
}

// ---------------- transformer aggregate: one block per node ----------------
__global__ void trans_aggregate(const float* __restrict__ vt, const float* __restrict__ ea,
                                const float* __restrict__ tew, const float* __restrict__ l2,
                                float* __restrict__ tout)
{
  int n = blockIdx.x, c = threadIdx.x;
  int g = n / 16, t = n % 16;
  __shared__ float w[HH][15];
  if (c < 120) {
    int si = c % 15, hh = c / 15;
    w[hh][si] = l2[((long)n * HH + hh) * 16 + si];
  }
  __syncthreads();
  if (c < HH) {
    float m = -1e30f;
    for (int i = 0; i < 15; ++i) m = fmaxf(m, w[c][i]);
    float ssum = 0.f;
    for (int i = 0; i < 15; ++i) { float ev = __expf(w[c][i] - m); w[c][i] = ev; ssum += ev; }
    float inv = 1.0f / fmaxf(ssum, 1e-16f);
    for (int i = 0; i < 15; ++i) w[c][i] *= inv;
  }
  __syncthreads();
  float acc = 0.f;
  for (int si = 0; si < 15; ++si) {
    int s = si + (si >= t ? 1 : 0);
    long src = g * 16 + s;
    int e = g * EPG + s * 15 + (s < t ? t - 1 : t);
    #pragma unroll
    for (int hh = 0; hh < HH; ++hh) {
      int hc = hh * CCd + c;
      float te = 0.f;
      #pragma unroll
      for (int j = 0; j < 6; ++j) te += ea[e * 6 + j] * tew[hc * 6 + j];
      acc += (vt[src * HCc + hc] + te) * w[hh][si];
    }
  }
  tout[n * DD + c] = acc * 0.125f;
}

// ---------------- beta gate + LN ----------------
__global__ void beta_combine_ln(const float* __restrict__ tout, const float* __restrict__ rskip,
                                const float* __restrict__ tbw, const float* __restrict__ lng,
                                const float* __restrict__ lnb, float* __restrict__ xtr)
{
  int n = blockIdx.x, d = threadIdx.x;
  float o = tout[n * DD + d], r = rskip[n * DD + d];
  float part = o * tbw[d] + r * tbw[DD + d] + (o - r) * tbw[2 * DD + d];
  __shared__ float sh[256];
  float z = block_sum(part, sh);
  float beta = 1.0f / (1.0f + __expf(-z));
  float xv = beta * r + (1.0f - beta) * o;
  float mean = block_sum(xv, sh) * (1.0f / DD);
  float dv = xv - mean;
  float var = block_sum(dv * dv, sh) * (1.0f / DD);
  xtr[n * DD + d] = dv * rsqrtf(var + 1e-5f) * lng[d] + lnb[d];
}

// ---------------- attention row softmax (mask is row-constant => dropped) ----------------
__global__ void softmax_row(float* __restrict__ attn)
{
  int i = blockIdx.x, hh = blockIdx.y, t = threadIdx.x;
  float* row = attn + ((long)hh * NN + i) * NN;
  float a = row[t], b = row[t + 256];
  __shared__ float sh[256];
  float m = block_max(fmaxf(a, b), sh);
  a = __expf(a - m); b = __expf(b - m);
  float ssum = block_sum(a + b, sh);
  float inv = 1.0f / ssum;
  row[t] = a * inv; row[t + 256] = b * inv;
}

// ---------------- pooling (softmax over axis of size 1 == 1) ----------------
__global__ void pool_kernel(const float* __restrict__ obuf, const float* __restrict__ hres,
                            float* __restrict__ pooled)
{
  int g = blockIdx.x, d = threadIdx.x;
  float osum = 0.f;
  for (int i = 0; i < NN; ++i) osum += obuf[(long)i * DD + d];
  float rsum = 0.f;
  for (int j = 0; j < 16; ++j) rsum += hres[(long)(g * 16 + j) * DD + d];
  pooled[g * DD + d] = 16.0f * osum + 512.0f * rsum;
}

// ---------------- LN + exact GELU (+ optional residual add) ----------------
__global__ void ln_gelu(const float* __restrict__ in, const float* __restrict__ g,
                        const float* __restrict__ b, float* __restrict__ out,
                        const float* __restrict__ addres)
{
  int n = blockIdx.x, d = threadIdx.x;
  float xv = in[n * 128 + d];
  __shared__ float sh[256];
  float mean = block_sum(xv, sh) * (1.0f / 128.0f);
  float dv = xv - mean;
  float var = block_sum(dv * dv, sh) * (1.0f / 128.0f);
  float y = dv * rsqrtf(var + 1e-5f) * g[d] + b[d];
  float gel = 0.5f * y * (1.0f + erff(y * 0.70710678118654752f));
  out[n * 128 + d] = (addres ? addres[n * 128 + d] : 0.0f) + gel;
}

// ---------------- final projection to (G,1) ----------------
__global__ void final_out(const float* __restrict__ c, const float* __restrict__ w,
                          const float* __restrict__ b, float* __restrict__ out)
{
  int g = blockIdx.x, d = threadIdx.x;
  __shared__ float sh[256];
  float tot = block_sum(c[g * 128 + d] * w[d], sh);
  if (d == 0) out[g] = tot + b[0];
}

extern "C" void kernel_launch(void* const* d_in, const int* in_sizes, int n_in,
                              void* d_out, int out_size, void* d_ws, size_t ws_size,
                              hipStream_t stream) {
  (void)in_sizes; (void)n_in; (void)out_size; (void)ws_size;
  const float* x        = (const float*)d_in[0];
  const float* ea       = (const float*)d_in[1];
  const float* emb_w    = (const float*)d_in[2];
  const float* emb_b    = (const float*)d_in[3];
  const float* emb_g    = (const float*)d_in[4];
  const float* emb_bb   = (const float*)d_in[5];
  const float* pe       = (const float*)d_in[6];
  const float* gat_lw   = (const float*)d_in[7];
  const float* gat_lb   = (const float*)d_in[8];
  const float* gat_rw   = (const float*)d_in[9];
  const float* gat_rb   = (const float*)d_in[10];
  const float* gat_ew   = (const float*)d_in[11];
  const float* gat_att  = (const float*)d_in[12];
  const float* gat_bias = (const float*)d_in[13];
  const float* tq_w     = (const float*)d_in[14];
  const float* tq_b     = (const float*)d_in[15];
  const float* tk_w     = (const float*)d_in[16];
  const float* tk_b     = (const float*)d_in[17];
  const float* tv_w     = (const float*)d_in[18];
  const float* tv_b     = (const float*)d_in[19];
  const float* te_w     = (const float*)d_in[20];
  const float* tskip_w  = (const float*)d_in[21];
  const float* tskip_b  = (const float*)d_in[22];
  const float* tbeta_w  = (const float*)d_in[23];
  const float* ln_g     = (const float*)d_in[24];
  const float* ln_b     = (const float*)d_in[25];
  const float* mha_in_w = (const float*)d_in[26];
  const float* mha_in_b = (const float*)d_in[27];
  const float* mha_ow   = (const float*)d_in[28];
  const float* mha_ob   = (const float*)d_in[29];
  const float* c1_w     = (const float*)d_in[32];
  const float* c1_b     = (const float*)d_in[33];
  const float* c1_g     = (const float*)d_in[34];
  const float* c1_bb    = (const float*)d_in[35];
  const float* rb_w     = (const float*)d_in[36];
  const float* rb_b     = (const float*)d_in[37];
  const float* rb_g     = (const float*)d_in[38];
  const float* rb_bb    = (const float*)d_in[39];
  const float* c2_w     = (const float*)d_in[40];
  const float* c2_b     = (const float*)d_in[41];

  float* WS = (float*)d_ws;                 // needs ~27 MB of scratch
  float* h      = WS + 0;                   // 512*256
  float* xl     = WS + 131072;              // 512*2048  (reused as qt)
  float* xr     = WS + 1179648;             // 512*2048  (reused as kt)
  float* vt     = WS + 2228224;             // 512*2048
  float* glog   = WS + 3276800;             // 7680*8
  float* xgat   = WS + 3338240;             // 512*256
  float* l2     = WS + 3469312;             // 512*8*16
  float* tout   = WS + 3534848;             // 512*256
  float* rskip  = WS + 3665920;             // 512*256
  float* xtr    = WS + 3796992;             // 512*256
  float* qf     = WS + 3928064;             // 512*256
  float* kf     = WS + 4059136;             // 512*256
  float* vf     = WS + 4190208;             // 512*256
  float* attn   = WS + 4321280;             // 8*512*512
  float* oh     = WS + 6418432;             // 512*256
  float* obuf   = WS + 6549504;             // 512*256
  float* pooled = WS + 6680576;             // 32*256
  float* t1     = WS + 6688768;             // 32*128
  float* cb     = WS + 6692864;             // 32*128
  float* t2     = WS + 6696960;             // 32*128
  float* cb2    = WS + 6701056;             // 32*128

  const dim3 gBig(HCc / 32, NN / 32);       // 512 x 2048 projections
  const dim3 gSq(DD / 32, NN / 32);         // 512 x 256 projections

  // 1) h = LN(x @ emb_w.T + emb_b) + pe
  embed_ln_pe<<<NN, 256, 0, stream>>>(x, emb_w, emb_b, emb_g, emb_bb, pe, h);

  // 2) GAT projections
  gemm_nt_wmma<<<gBig, 32, 0, stream>>>(h, gat_lw, gat_lb, xl, DD, DD, DD, HCc, 1.0f, 0, 0, 0);
  gemm_nt_wmma<<<gBig, 32, 0, stream>>>(h, gat_rw, gat_rb, xr, DD, DD, DD, HCc, 1.0f, 0, 0, 0);

  // 3) GAT edge logits + per-dst softmax aggregate + LN
  gat_edge_logits<<<dim3(EEe, HH), 256, 0, stream>>>(xl, xr, ea, gat_ew, gat_att, glog);
  gat_aggregate<<<NN, 256, 0, stream>>>(xl, glog, gat_bias, ln_g, ln_b, xgat);

  // 4) transformer projections (qt/kt reuse xl/xr scratch)
  float* qt = xl; float* kt = xr;
  gemm_nt_wmma<<<gBig, 32, 0, stream>>>(xgat, tq_w, tq_b, qt, DD, DD, DD, HCc, 1.0f, 0, 0, 0);
  gemm_nt_wmma<<<gBig, 32, 0, stream>>>(xgat, tk_w, tk_b, kt, DD, DD, DD, HCc, 1.0f, 0, 0, 0);
  gemm_nt_wmma<<<gBig, 32, 0, stream>>>(xgat, tv_w, tv_b, vt, DD, DD, DD, HCc, 1.0f, 0, 0, 0);

  // 5) transformer edge attention
  trans_logits<<<dim3(NN, HH), 256, 0, stream>>>(qt, kt, ea, te_w, l2);
  trans_aggregate<<<NN, 256, 0, stream>>>(vt, ea, te_w, l2, tout);

  // 6) gated residual
  gemm_nt_wmma<<<gSq, 32, 0, stream>>>(xgat, tskip_w, tskip_b, rskip, DD, DD, DD, DD, 1.0f, 0, 0, 0);
  beta_combine_ln<<<NN, 256, 0, stream>>>(tout, rskip, tbeta_w, ln_g, ln_b, xtr);

  // 7) MHA projections: q from x_tr, k from x_gat, v from h
  gemm_nt_wmma<<<gSq, 32, 0, stream>>>(xtr,  mha_in_w,            mha_in_b,       qf, DD, DD, DD, DD, 1.0f, 0, 0, 0);
  gemm_nt_wmma<<<gSq, 32, 0, stream>>>(xgat, mha_in_w + 256*256,  mha_in_b + 256, kf, DD, DD, DD, DD, 1.0f, 0, 0, 0);
  gemm_nt_wmma<<<gSq, 32, 0, stream>>>(h,    mha_in_w + 512*256,  mha_in_b + 512, vf, DD, DD, DD, DD, 1.0f, 0, 0, 0);

  // 8) attention: logits = (q/sqrt(HD)) @ k^T per head (mask is row-constant -> no-op)
  gemm_nt_wmma<<<dim3(NN / 32, NN / 32, HH), 32, 0, stream>>>(
      qf, kf, nullptr, attn, HD32, DD, DD, NN, 0.17677669529663689f,
      HD32, HD32, (long)NN * NN);
  softmax_row<<<dim3(NN, HH), 256, 0, stream>>>(attn);
  gemm_nn_wmma<<<dim3(HD32 / 32, NN / 32, HH), 32, 0, stream>>>(
      attn, vf, nullptr, oh, NN, NN, DD, DD, 1.0f,
      (long)NN * NN, HD32, HD32);
  gemm_nt_wmma<<<gSq, 32, 0, stream>>>(oh, mha_ow, mha_ob, obuf, DD, DD, DD, DD, 1.0f, 0, 0, 0);

  // 9) pooled[g] = 16*sum_i o[i] + 512*sum_{j in g} residual[j]   (score==1 identity)
  pool_kernel<<<GGg, 256, 0, stream>>>(obuf, h, pooled);

  // 10) classifier head
  gemm_nt_wmma<<<dim3(128 / 32, GGg / 32), 32, 0, stream>>>(pooled, c1_w, c1_b, t1, DD, DD, DD, 128, 1.0f, 0, 0, 0);
  ln_gelu<<<GGg, 128, 0, stream>>>(t1, c1_g, c1_bb, cb, nullptr);
  gemm_nt_wmma<<<dim3(128 / 32, GGg / 32), 32, 0, stream>>>(cb, rb_w, rb_b, t2, 128, 128, 128, 128, 1.0f, 0, 0, 0);
  ln_gelu<<<GGg, 128, 0, stream>>>(t2, rb_g, rb_bb, cb2, cb);
  final_out<<<GGg, 128, 0, stream>>>(cb2, c2_w, c2_b, (float*)d_out);
}